// BaseFullTensorProduct_33595234189529
// MI455X (gfx1250) — compile-verified
//
#include <hip/hip_runtime.h>
#include <math.h>

// ============================================================================
// e3nn full tensor product, IRREPS1=[(64,0,+),(64,1,-),(32,2,+)] x IRREPS2=[(1,0,+),(1,1,-),(1,2,+)]
// N=50000, DIM1=416, DIM2=9, DIM_OUT=3744.
//
// Roofline (MI455X): traffic = 83MB(x1)+2MB(x2)+749MB(out) ~ 834MB -> ~36us @23.3TB/s.
// Factored form (mul2==1): y[u,k] = sum_i x1[u,i]*W[i,k], W = CG . x2 => 1.3 GFLOP: bandwidth-bound.
// Per-row tiny matmuls mapped onto V_WMMA_F32_16X16X4_F32 (M=16 mults, N=16 cols, K=d1 in {1,3,5}):
// 24 WMMAs/row, 1.2M total -> compute is free. Key tuning concern: VGPR pressure (occupancy
// for HBM latency hiding). Store offsets are made OPAQUE via empty inline asm so LICM cannot
// pre-materialize 24x8 hoisted 64-bit offsets; they are recomputed per row as cheap 32-bit chains.
// ============================================================================

typedef __attribute__((ext_vector_type(2))) float v2f;
typedef __attribute__((ext_vector_type(8))) float v8f;

#define NWAVES 8
#define BMAT_FLOATS 2835   // W-builder: block0 1*9*9=81, block1 3*27*9=729, block2 5*45*9=2025
#define WV_FLOATS 576      // padded per-wave W: 4x16 + 4x32 + 8x48

#if __has_builtin(__builtin_amdgcn_sched_barrier)
#define SCHED_FENCE() __builtin_amdgcn_sched_barrier(0)
#else
#define SCHED_FENCE()
#endif

// ---------------- output-ordering tables (stable sort by (l3, -p*(-1)^l3)) ----------------
__device__ static constexpr int CS0[3]  = {0,1,4};
__device__ static constexpr int D3T0[3] = {1,3,5};
__device__ static constexpr int OF0[3]  = {0,160,1120};
__device__ static constexpr int CS1[7]  = {0,3,4,7,12,15,20};
__device__ static constexpr int D3T1[7] = {3,1,3,5,3,5,7};
__device__ static constexpr int OF1[7]  = {352,64,832,1440,544,2080,2560};
__device__ static constexpr int CS2[9]  = {0,5,8,13,20,21,24,29,36};
__device__ static constexpr int D3T2[9] = {5,3,5,7,1,3,5,7,9};
__device__ static constexpr int OF2[9]  = {1760,736,2400,3008,128,1024,1920,3232,3456};

// ---------------- init-kernel path tables ----------------
__device__ static constexpr int P_L1[19] = {0,0,0, 1,1,1,1,1,1,1, 2,2,2,2,2,2,2,2,2};
__device__ static constexpr int P_L2[19] = {0,1,2, 0,1,1,1,2,2,2, 0,1,1,1,2,2,2,2,2};
__device__ static constexpr int P_L3[19] = {0,1,2, 1,0,1,2,1,2,3, 2,1,2,3,0,1,2,3,4};
__device__ static constexpr int P_CS[19] = {0,1,4, 0,3,4,7,12,15,20, 0,5,8,13,20,21,24,29,36};
__device__ static constexpr int BLK_BASE[3] = {0,81,810};
__device__ static constexpr int BLK_D3[3]   = {9,27,45};
__device__ static constexpr int S2T[3]      = {0,1,4};   // x2 offset per l2

// ============================================================================
// CG init: exact e3nn Clebsch-Gordan (real basis, unit Frobenius, * sqrt(2l3+1)),
// folded into W-builder matrices B[(i,col),j] so that W[i,col] = sum_j B*x2[j].
// Runs every launch (deterministic, ~us). One block per path.
// ============================================================================
__device__ static double dfact(int n) { double r = 1.0; for (int i = 2; i <= n; ++i) r *= (double)i; return r; }

__device__ static double su2_cg(int j1, int j2, int j3, int m1, int m2) {
  int m3 = m1 + m2;
  int vmin = -j1 + j2 + m3; if (-j1 + m1 > vmin) vmin = -j1 + m1; if (vmin < 0) vmin = 0;
  int vmax = j2 + j3 + m1; if (j3 - j1 + j2 < vmax) vmax = j3 - j1 + j2; if (j3 + m3 < vmax) vmax = j3 + m3;
  double C = (double)(2 * j3 + 1) *
      (dfact(j3 + j1 - j2) * dfact(j3 - j1 + j2) * dfact(j1 + j2 - j3) * dfact(j3 + m3) * dfact(j3 - m3)) /
      (dfact(j1 + j2 + j3 + 1) * dfact(j1 - m1) * dfact(j1 + m1) * dfact(j2 - m2) * dfact(j2 + m2));
  C = sqrt(C);
  double S = 0.0;
  for (int v = vmin; v <= vmax; ++v) {
    double sgn = ((v + j2 + m2) & 1) ? -1.0 : 1.0;
    S += sgn * (dfact(j2 + j3 + m1 - v) * dfact(j1 - m1 + v)) /
         (dfact(v) * dfact(j3 - j1 + j2 - v) * dfact(j3 + m3 - v) * dfact(v + j1 - j2 - m3));
  }
  return C * S;
}

// real->complex spherical harmonics change of basis (e3nn convention), times (-i)^l
__device__ static void build_q(int l, double* qr, double* qi) {
  int d = 2 * l + 1;
  for (int i = 0; i < d * d; ++i) { qr[i] = 0.0; qi[i] = 0.0; }
  const double is2 = 0.70710678118654752440;
  for (int m = 1; m <= l; ++m) {
    qr[(l - m) * d + (l + m)] = is2;
    qi[(l - m) * d + (l - m)] = -is2;
    double s = (m & 1) ? -1.0 : 1.0;
    qr[(l + m) * d + (l + m)] = s * is2;
    qi[(l + m) * d + (l - m)] = s * is2;
  }
  qr[l * d + l] = 1.0;
  int ph = l & 3;                              // multiply by (-i)^l
  if (ph) {
    for (int i = 0; i < d * d; ++i) {
      double re = qr[i], im = qi[i];
      if (ph == 1)      { qr[i] =  im; qi[i] = -re; }
      else if (ph == 2) { qr[i] = -re; qi[i] = -im; }
      else              { qr[i] = -im; qi[i] =  re; }
    }
  }
}

__global__ void cg_init_kernel(float* __restrict__ Bmat) {
  __shared__ double q1r[81], q1i[81], q2r[81], q2i[81], q3r[81], q3i[81];
  __shared__ double csu2[225];
  __shared__ double cgout[225];
  __shared__ double scale_sh;
  const int p = blockIdx.x;
  const int l1 = P_L1[p], l2 = P_L2[p], l3 = P_L3[p];
  const int d1 = 2 * l1 + 1, d2 = 2 * l2 + 1, d3 = 2 * l3 + 1;
  const int t = threadIdx.x;

  if (t == 0) { build_q(l1, q1r, q1i); build_q(l2, q2r, q2i); build_q(l3, q3r, q3i); }
  for (int idx = t; idx < d1 * d2 * d3; idx += blockDim.x) {
    int i = idx / (d2 * d3), rem = idx % (d2 * d3), j = rem / d3, k = rem % d3;
    int m1 = i - l1, m2 = j - l2, ms = m1 + m2;
    int msa = ms < 0 ? -ms : ms;
    csu2[idx] = (msa <= l3 && k == l3 + ms) ? su2_cg(l1, l2, l3, m1, m2) : 0.0;
  }
  __syncthreads();

  // real CG: cg[a,b,c] = Re( sum_{i,k,n} q1[i,a] q2[k,b] conj(q3[n,c]) csu2[i,k,n] )
  for (int idx = t; idx < d1 * d2 * d3; idx += blockDim.x) {
    int a = idx / (d2 * d3), rem = idx % (d2 * d3), b = rem / d3, c = rem % d3;
    double sre = 0.0;
    for (int i = 0; i < d1; ++i) {
      double a1r = q1r[i * d1 + a], a1i = q1i[i * d1 + a];
      if (a1r == 0.0 && a1i == 0.0) continue;
      for (int k = 0; k < d2; ++k) {
        double a2r = q2r[k * d2 + b], a2i = q2i[k * d2 + b];
        if (a2r == 0.0 && a2i == 0.0) continue;
        double t12r = a1r * a2r - a1i * a2i;
        double t12i = a1r * a2i + a1i * a2r;
        for (int n = 0; n < d3; ++n) {
          double cv = csu2[(i * d2 + k) * d3 + n];
          if (cv == 0.0) continue;
          double a3r = q3r[n * d3 + c], a3i = -q3i[n * d3 + c];  // conj
          sre += cv * (t12r * a3r - t12i * a3i);
        }
      }
    }
    cgout[idx] = sre;
  }
  __syncthreads();
  if (t == 0) {
    double ss = 0.0;
    for (int i = 0; i < d1 * d2 * d3; ++i) ss += cgout[i] * cgout[i];
    scale_sh = sqrt((double)(2 * l3 + 1)) / sqrt(ss);
  }
  __syncthreads();
  const double sc = scale_sh;
  const int D3b = BLK_D3[l1], base = BLK_BASE[l1], cs = P_CS[p], s2 = S2T[l2];
  for (int idx = t; idx < d1 * d3 * 9; idx += blockDim.x) {
    int a = idx / (d3 * 9), rem = idx % (d3 * 9), c = rem / 9, j = rem % 9;
    float val = 0.f;
    if (j >= s2 && j < s2 + d2) val = (float)(cgout[(a * d2 + (j - s2)) * d3 + c] * sc);
    Bmat[base + (a * D3b + cs + c) * 9 + j] = val;
  }
}

// ============================================================================
// Main kernel: one row per wave per iteration, V_WMMA_F32_16X16X4_F32 tiles.
// A layout (16x4 f32): lanes 0-15 hold M=0..15 K={0,1}, lanes 16-31 K={2,3}.
// B layout mirror: lane%16 = N col; VGPR0 = K rows {0|2}, VGPR1 = {1|3}.
// D layout: lanes 0-15: N=lane, VGPR r -> M=r; lanes 16-31: N=lane-16, M=r+8.
// ============================================================================

// per-lane column -> (absolute chunk offset, u-stride) map, computed ONCE per lane
template <int NPATH>
__device__ __forceinline__ void colmap(int col, const int (&cs)[NPATH], const int (&d3t)[NPATH],
                                       const int (&oft)[NPATH], int& off, int& str) {
  off = 0; str = 0;
#pragma unroll
  for (int t = 0; t < NPATH; ++t)
    if (col >= cs[t] && col < cs[t] + d3t[t]) { off = oft[t] + (col - cs[t]); str = d3t[t]; }
}

template <int D1, int D3BLK, int WPAD, int MULS, int KCH>
__device__ __forceinline__ void do_block(const float* __restrict__ x1b,
                                         float* __restrict__ outrow,
                                         const float* __restrict__ wv,
                                         const int* offA, const int* strA,  // per-tile, per-lane
                                         int lane) {
  const int m = lane & 15;
  const int kh = lane >> 4;
  constexpr int NT = (D3BLK + 15) / 16;
  constexpr int MT = MULS / 16;
#pragma unroll
  for (int nt = 0; nt < NT; ++nt) {
    const int col = nt * 16 + m;
    // folds to compile-time 'true' for full tiles after unrolling
    const bool cvalid = ((nt + 1) * 16 <= D3BLK) || (col < D3BLK);
    v2f b;                               // unconditional: W buffer is tile-padded with zeros
    b.x = wv[(2 * kh + 0) * WPAD + col];
    b.y = wv[(2 * kh + 1) * WPAD + col];
    v2f b2; b2.x = 0.f; b2.y = 0.f;
    if (KCH == 2) {
      b2.x = wv[(4 + 2 * kh) * WPAD + col];
      b2.y = wv[(5 + 2 * kh) * WPAD + col];
    }
    int off = offA[nt];                  // constant index after unroll -> register
    int str = strA[nt];
    // OPTIMIZATION BARRIER: make store-offset roots opaque so LICM cannot hoist
    // 24 tiles x 8 offsets out of the row loop (the round-2/3 VGPR-pressure bug).
    __asm__ volatile("" : "+v"(off), "+v"(str));
#pragma unroll
    for (int mt = 0; mt < MT; ++mt) {
      const int u = mt * 16 + m;
      v2f a;
      if (D1 == 1) {                      // K row 0 only; upper half-wave holds zeros
        float v0 = x1b[u];
        a.x = kh ? 0.f : v0; a.y = 0.f;
      } else if (D1 == 3) {               // K rows 0..2; offset-select keeps loads unconditional
        const float* p = x1b + u * 3 + (kh ? 2 : 0);
        float v0 = p[0], v1 = p[1];       // p[1] for kh lanes reads in-bounds junk, masked below
        a.x = v0; a.y = kh ? 0.f : v1;
      } else {                            // D1 == 5, K rows 0..3 in this chunk
        const float* p = x1b + u * 5 + (kh ? 2 : 0);
        a.x = p[0]; a.y = p[1];
      }
      v8f acc = {};
      acc = __builtin_amdgcn_wmma_f32_16x16x4_f32(false, a, false, b, (short)0, acc, false, false);
      if (KCH == 2) {                     // K=4 tail for d1==5 (rows 5..7 zero-padded in LDS)
        float v4 = x1b[u * 5 + 4];
        v2f a2; a2.x = kh ? 0.f : v4; a2.y = 0.f;
        acc = __builtin_amdgcn_wmma_f32_16x16x4_f32(false, a2, false, b2, (short)0, acc, false, false);
      }
      if (cvalid) {
        // serial 32-bit chain off the opaque root: one live offset,
        // v_add_nc_u32 between saddr-form stores
        unsigned o = (unsigned)(off + (mt * 16 + kh * 8) * str);
        const unsigned st = (unsigned)str;
#pragma unroll
        for (int r = 0; r < 8; ++r) { outrow[o] = acc[r]; o += st; }
      }
    }
  }
}

__global__ __launch_bounds__(256) void tp_wmma_kernel(const float* __restrict__ x1,
                                                      const float* __restrict__ x2,
                                                      float* __restrict__ out,
                                                      const float* __restrict__ Bmat, int n) {
  __shared__ float sB[BMAT_FLOATS];        // W-builder, cached in LDS (11.3 KB)
  __shared__ float sW[NWAVES * WV_FLOATS]; // per-wave W, tile-padded: 4x16 + 4x32 + 8x48
  for (int i = threadIdx.x; i < BMAT_FLOATS; i += blockDim.x) sB[i] = Bmat[i];
  __syncthreads();

  const int lane = threadIdx.x & 31;
  const int wave = threadIdx.x >> 5;
  const int m = lane & 15;
  float* wv = &sW[wave * WV_FLOATS];
  const int stride = gridDim.x * NWAVES;

  // row-invariant per-lane store maps for the 6 column-tiles (12 live ints total)
  int offA[6], strA[6];
  colmap<3>(m,      CS0, D3T0, OF0, offA[0], strA[0]);   // blk0 tile0 (cols 0..8)
  colmap<7>(m,      CS1, D3T1, OF1, offA[1], strA[1]);   // blk1 tile0
  colmap<7>(16 + m, CS1, D3T1, OF1, offA[2], strA[2]);   // blk1 tile1 (cols 16..26)
  colmap<9>(m,      CS2, D3T2, OF2, offA[3], strA[3]);   // blk2 tile0
  colmap<9>(16 + m, CS2, D3T2, OF2, offA[4], strA[4]);   // blk2 tile1
  colmap<9>(32 + m, CS2, D3T2, OF2, offA[5], strA[5]);   // blk2 tile2 (cols 32..44)

  for (int row = blockIdx.x * NWAVES + wave; row < n; row += stride) {
    const int r = __builtin_amdgcn_readfirstlane(row);   // force scalar bases
    const float* x1row = x1 + (long)r * 416;
    const float* x2row = x2 + (long)r * 9;
    float* outrow = out + (long)r * 3744;

    float xv[9];
#pragma unroll
    for (int j = 0; j < 9; ++j) xv[j] = x2row[j];

    // build padded W for this row, distributed over the 32 lanes (18 slots each)
    for (int s = lane; s < WV_FLOATS; s += 32) {
      int krow, col, d1v, D3v, base;
      if (s < 64)       { krow = s >> 4;          col = s & 15;  d1v = 1; D3v = 9;  base = 0;   }
      else if (s < 192) { int q = s - 64;  krow = q >> 5; col = q & 31;  d1v = 3; D3v = 27; base = 81;  }
      else              { int q = s - 192; krow = q / 48; col = q % 48; d1v = 5; D3v = 45; base = 810; }
      float v = 0.f;
      if (krow < d1v && col < D3v) {
        const float* bm = &sB[base + (krow * D3v + col) * 9];
#pragma unroll
        for (int j = 0; j < 9; ++j) v = fmaf(bm[j], xv[j], v);
      }
      wv[s] = v;
    }
    // wave-private LDS exchange: drain this wave's DS ops (CDNA5 split counter)
    __asm__ volatile("s_wait_dscnt 0x0" ::: "memory");

    do_block<1, 9, 16, 64, 1>(x1row,        outrow, wv,       &offA[0], &strA[0], lane);  // 4 WMMA
    SCHED_FENCE();
    do_block<3, 27, 32, 64, 1>(x1row + 64,  outrow, wv + 64,  &offA[1], &strA[1], lane);  // 8 WMMA
    SCHED_FENCE();
    do_block<5, 45, 48, 32, 2>(x1row + 256, outrow, wv + 192, &offA[3], &strA[3], lane);  // 12 WMMA
    SCHED_FENCE();
  }
}

// ============================================================================
extern "C" void kernel_launch(void* const* d_in, const int* in_sizes, int n_in,
                              void* d_out, int out_size, void* d_ws, size_t ws_size,
                              hipStream_t stream) {
  const float* x1 = (const float*)d_in[0];
  const float* x2 = (const float*)d_in[1];
  float* out = (float*)d_out;
  float* Bmat = (float*)d_ws;                 // 2835 floats of workspace
  const int n = in_sizes[0] / 416;

  cg_init_kernel<<<19, 64, 0, stream>>>(Bmat);

  int blocks = (n + NWAVES - 1) / NWAVES;
  if (blocks > 1280) blocks = 1280;           // grid-stride; amortizes LDS staging of Bmat
  tp_wmma_kernel<<<blocks, 256, 0, stream>>>(x1, x2, out, Bmat, n);
}